// RNN_16492674416646
// MI455X (gfx1250) — compile-verified
//
#include <hip/hip_runtime.h>

typedef float v2f __attribute__((ext_vector_type(2)));
typedef float v8f __attribute__((ext_vector_type(8)));

namespace {
constexpr int kT  = 2048;
constexpr int kB  = 128;
constexpr int kH  = 200;
constexpr int kHS = 204;                       // h row stride (floats): 200 h | in0 | in1 | 1.0 | 0pad
constexpr int kKB = 51;                        // K blocks of 4 (K padded 200 -> 204)
constexpr int kWBR = 102;                      // float2 rows of B matrix (204/2)
constexpr int kWBC = 208;                      // padded N columns (13 * 16)
constexpr int kWBBytes  = kWBR * kWBC * 8;     // 169728
constexpr int kHBytes   = kB * kHS * 4;        // 104448
constexpr int kLdsBytes = kWBBytes + kHBytes;  // 274176 <= 320K WGP LDS
}

__global__ __launch_bounds__(1024)
void rnn_fused_wmma(const float* __restrict__ in_seq,  // [T][B][2]
                    const float* __restrict__ W_ih,    // [200][2]
                    const float* __restrict__ W_hh,    // [200][200]
                    const float* __restrict__ b_ih,    // [200]
                    const float* __restrict__ b_hh,    // [200]
                    const float* __restrict__ W_out,   // [1][200]
                    const float* __restrict__ b_out,   // [1]
                    float* __restrict__ out)           // [T][B][1]
{
  extern __shared__ char lds[];
  v2f*   wb = reinterpret_cast<v2f*>(lds);               // B matrix: [102][208] float2 {W[k][j], W[k+1][j]}
  float* hb = reinterpret_cast<float*>(lds + kWBBytes);  // h state:  [128][204]

  const int tid  = threadIdx.x;
  const int lane = tid & 31;
  const int wave = tid >> 5;

  // ---------------- one-time init ----------------
  // B-operand matrix: Bext[k][j] = W_hh[j][k] for k<200; rows 200,201 = W_ih cols; row 202 = biases; row 203 = 0
  for (int idx = tid; idx < kWBR * kWBC; idx += 1024) {
    const int k2 = idx / kWBC;
    const int j  = idx - k2 * kWBC;
    v2f v = {0.0f, 0.0f};
    if (j < kH) {
      if (k2 < kH / 2) {
        v = v2f{W_hh[j * kH + 2 * k2], W_hh[j * kH + 2 * k2 + 1]};
      } else if (k2 == kH / 2) {
        v = v2f{W_ih[j * 2 + 0], W_ih[j * 2 + 1]};
      } else {
        v = v2f{b_ih[j] + b_hh[j], 0.0f};
      }
    }
    wb[idx] = v;
  }
  // h buffer: h0 = 0; col 202 = 1.0 (bias lane), col 203 = 0 pad
  for (int idx = tid; idx < kB * kHS; idx += 1024) {
    const int c = idx % kHS;
    hb[idx] = (c == 202) ? 1.0f : 0.0f;
  }
  __syncthreads();
  // stage in[0] into cols 200..201
  if (tid < kB) {
    const v2f x0 = reinterpret_cast<const v2f*>(in_seq)[tid];
    *reinterpret_cast<v2f*>(&hb[tid * kHS + 200]) = x0;
  }

  // head weights in registers: thread tid -> (batch row bhead, 25-wide chunk)
  const int chunk = tid & 7;
  const int bhead = tid >> 3;
  float wout[25];
#pragma unroll
  for (int i = 0; i < 25; ++i) wout[i] = W_out[chunk * 25 + i];
  const float bout = b_out[0];

  // GEMM tile assignment: 26 waves; wave -> (nT = wave>>1, mGroup = wave&1), 4 M-tiles each
  const bool gemmWave = (wave < 26);
  const int nT     = wave >> 1;
  const int mGroup = wave & 1;
  const int col    = nT * 16 + (lane & 15);
  int aOff[4];
#pragma unroll
  for (int i = 0; i < 4; ++i)
    aOff[i] = ((4 * mGroup + i) * 16 + (lane & 15)) * kHS + ((lane >> 4) << 1);
  const int bOff0 = (lane >> 4) * kWBC + col;

  __syncthreads();

  const v8f vzero = {0.f, 0.f, 0.f, 0.f, 0.f, 0.f, 0.f, 0.f};

  // ---------------- serial recurrence ----------------
#pragma unroll 1
  for (int t = 0; t < kT; ++t) {
    v8f acc[4];
#pragma unroll
    for (int i = 0; i < 4; ++i) acc[i] = vzero;

    if (gemmWave) {
      int bo = bOff0;
      for (int kb = 0; kb < kKB; ++kb) {
        const v2f bop = wb[bo];
        const v2f a0 = *reinterpret_cast<const v2f*>(&hb[aOff[0] + (kb << 2)]);
        const v2f a1 = *reinterpret_cast<const v2f*>(&hb[aOff[1] + (kb << 2)]);
        const v2f a2 = *reinterpret_cast<const v2f*>(&hb[aOff[2] + (kb << 2)]);
        const v2f a3 = *reinterpret_cast<const v2f*>(&hb[aOff[3] + (kb << 2)]);
        acc[0] = __builtin_amdgcn_wmma_f32_16x16x4_f32(false, a0, false, bop, (short)0, acc[0], false, false);
        acc[1] = __builtin_amdgcn_wmma_f32_16x16x4_f32(false, a1, false, bop, (short)0, acc[1], false, false);
        acc[2] = __builtin_amdgcn_wmma_f32_16x16x4_f32(false, a2, false, bop, (short)0, acc[2], false, false);
        acc[3] = __builtin_amdgcn_wmma_f32_16x16x4_f32(false, a3, false, bop, (short)0, acc[3], false, false);
        bo += 2 * kWBC;
      }
    }
    __syncthreads();  // all reads of h_{t-1} (and staged in[t]) complete

    if (gemmWave) {
      const bool colOk = (col < kH);
#pragma unroll
      for (int i = 0; i < 4; ++i) {
        const int rbase = (4 * mGroup + i) * 16 + ((lane >> 4) << 3);
#pragma unroll
        for (int v = 0; v < 8; ++v) {
          const float x  = acc[i][v];
          // tanh(x) = 1 - 2/(exp2(2x*log2e)+1); saturates correctly at +-1
          const float e  = __builtin_amdgcn_exp2f(x * 2.885390081777927f);
          const float th = 1.0f - 2.0f * __builtin_amdgcn_rcpf(e + 1.0f);
          if (colOk) hb[(rbase + v) * kHS + col] = th;
        }
      }
    } else {
      // staging waves: write in[t+1] into h cols 200..201
      const int b = tid - 832;
      if (b < kB && (t + 1) < kT) {
        const v2f x = reinterpret_cast<const v2f*>(in_seq)[(t + 1) * kB + b];
        *reinterpret_cast<v2f*>(&hb[b * kHS + 200]) = x;
      }
    }
    __syncthreads();  // h_t visible to all waves

    // output head: out[t][b] = dot(h_t[b,:200], W_out) + b_out
    float p = 0.0f;
    const float* hrow = &hb[bhead * kHS + chunk * 25];
#pragma unroll
    for (int i = 0; i < 25; ++i) p += hrow[i] * wout[i];
    p += __shfl_xor(p, 1, 32);
    p += __shfl_xor(p, 2, 32);
    p += __shfl_xor(p, 4, 32);
    if (chunk == 0) out[t * kB + bhead] = p + bout;
  }
}

extern "C" void kernel_launch(void* const* d_in, const int* in_sizes, int n_in,
                              void* d_out, int out_size, void* d_ws, size_t ws_size,
                              hipStream_t stream) {
  const float* in_seq = (const float*)d_in[0];
  const float* W_ih   = (const float*)d_in[1];
  const float* W_hh   = (const float*)d_in[2];
  const float* b_ih   = (const float*)d_in[3];
  const float* b_hh   = (const float*)d_in[4];
  const float* W_out  = (const float*)d_in[5];
  const float* b_out  = (const float*)d_in[6];
  float* out = (float*)d_out;

  hipLaunchKernelGGL(rnn_fused_wmma, dim3(1), dim3(1024), kLdsBytes, stream,
                     in_seq, W_ih, W_hh, b_ih, b_hh, W_out, b_out, out);
}